// GAT_40424232190065
// MI455X (gfx1250) — compile-verified
//
#include <hip/hip_runtime.h>
#include <hip/hip_bf16.h>
#include <math.h>

typedef __attribute__((ext_vector_type(2))) float v2f;
typedef __attribute__((ext_vector_type(8))) float v8f;

// ---------- helpers: order-preserving float<->uint encoding for atomicMax ----
__device__ __forceinline__ unsigned enc_f32(float f) {
  unsigned u = __float_as_uint(f);
  return (u & 0x80000000u) ? ~u : (u | 0x80000000u);
}
__device__ __forceinline__ float dec_f32(unsigned u) {
  unsigned b = (u & 0x80000000u) ? (u & 0x7FFFFFFFu) : ~u;
  return __uint_as_float(b);
}
__device__ __forceinline__ float leaky(float x) { return x >= 0.f ? x : 0.2f * x; }

// ---------- zero init ----------
__global__ void gat_zero_u32(unsigned* __restrict__ p, long long n) {
  long long i = (long long)blockIdx.x * blockDim.x + threadIdx.x;
  if (i < n) p[i] = 0u;
}

// ---------- GEMM1: C[M,256] = A[M,256] * B[256,256]; M % 16 == 0 ----------
// block = 256 threads = 8 waves; wave w computes 16x32 tile (two 16x16 wmma accs)
__global__ void gat_gemm_wmma_256x256(const float* __restrict__ A,
                                      const float* __restrict__ B,
                                      float* __restrict__ C) {
  const int lane = threadIdx.x & 31;
  const int wave = threadIdx.x >> 5;
  const int m0 = blockIdx.x << 4;
  const int n0 = wave << 5;          // 8 waves * 32 cols = 256 cols
  const int r = lane & 15;
  const int koff = (lane >> 4) << 1; // lanes 0-15 -> K 0,1 ; lanes 16-31 -> K 2,3
  v8f acc0 = {};
  v8f acc1 = {};
  const float* Arow = A + (size_t)(m0 + r) * 256 + koff;
  for (int k = 0; k < 256; k += 4) {
    v2f a;
    a.x = Arow[k];
    a.y = Arow[k + 1];
    const float* B0 = B + (size_t)(k + koff) * 256 + n0 + r;
    v2f b0, b1;
    b0.x = B0[0];  b0.y = B0[256];
    b1.x = B0[16]; b1.y = B0[256 + 16];
    acc0 = __builtin_amdgcn_wmma_f32_16x16x4_f32(false, a, false, b0, (short)0, acc0, false, false);
    acc1 = __builtin_amdgcn_wmma_f32_16x16x4_f32(false, a, false, b1, (short)0, acc1, false, false);
  }
  const int rbase = (lane >> 4) << 3; // C/D: vgpr v holds row v (lanes 0-15) / v+8 (lanes 16-31)
  for (int v = 0; v < 8; ++v) {
    C[(size_t)(m0 + rbase + v) * 256 + n0 + r]      = acc0[v];
    C[(size_t)(m0 + rbase + v) * 256 + n0 + 16 + r] = acc1[v];
  }
}

// ---------- GEMM2: C[M,16] = A[M,256] * B[256,16] ----------
// block = 256 = 8 waves; each wave owns one 16-row tile
__global__ void gat_gemm_wmma_256x16(const float* __restrict__ A,
                                     const float* __restrict__ B,
                                     float* __restrict__ C, int Mtiles) {
  const int lane = threadIdx.x & 31;
  const int wave = threadIdx.x >> 5;
  const int tile = blockIdx.x * 8 + wave;
  if (tile >= Mtiles) return;        // wave-uniform: EXEC stays all-ones for wmma
  const int m0 = tile << 4;
  const int r = lane & 15;
  const int koff = (lane >> 4) << 1;
  v8f acc = {};
  const float* Arow = A + (size_t)(m0 + r) * 256 + koff;
  for (int k = 0; k < 256; k += 4) {
    v2f a;
    a.x = Arow[k];
    a.y = Arow[k + 1];
    const float* B0 = B + (size_t)(k + koff) * 16 + r;
    v2f b;
    b.x = B0[0];
    b.y = B0[16];
    acc = __builtin_amdgcn_wmma_f32_16x16x4_f32(false, a, false, b, (short)0, acc, false, false);
  }
  const int rbase = (lane >> 4) << 3;
  for (int v = 0; v < 8; ++v)
    C[(size_t)(m0 + rbase + v) * 16 + r] = acc[v];
}

// ---------- per-node attention logits, layer 1 (8 heads x 32 dims) ----------
__global__ void gat_al1(const float* __restrict__ H, const float* __restrict__ asrc,
                        const float* __restrict__ adst, float* __restrict__ als,
                        float* __restrict__ ald, long long N8) {
  long long i = (long long)blockIdx.x * blockDim.x + threadIdx.x;
  if (i >= N8) return;
  long long n = i >> 3;
  int h = (int)(i & 7);
  const float* hp = H + n * 256 + h * 32;
  const float* sp = asrc + h * 32;
  const float* dp = adst + h * 32;
  float vs = 0.f, vd = 0.f;
  for (int c = 0; c < 32; ++c) {
    float x = hp[c];
    vs += x * sp[c];
    vd += x * dp[c];
  }
  als[i] = vs;
  ald[i] = vd;
}

// ---------- layer-1 edge max ----------
__global__ void gat_edge1_max(const long long* __restrict__ ei, long long E, long long Etot,
                              const float* __restrict__ als, const float* __restrict__ ald,
                              unsigned* __restrict__ menc) {
  long long e = (long long)blockIdx.x * blockDim.x + threadIdx.x;
  if (e >= Etot) return;
  long long s, d;
  if (e < E) { s = ei[e]; d = ei[E + e]; } else { s = d = e - E; }
  for (int h = 0; h < 8; ++h) {
    float sc = leaky(als[s * 8 + h] + ald[d * 8 + h]);
    atomicMax(&menc[d * 8 + h], enc_f32(sc));
  }
}

// ---------- decode encoded max in place ----------
__global__ void gat_dec_max(unsigned* __restrict__ m, long long n) {
  long long i = (long long)blockIdx.x * blockDim.x + threadIdx.x;
  if (i >= n) return;
  float f = dec_f32(m[i]);
  ((float*)m)[i] = f;
}

// ---------- layer-1 exp + denom ----------
__global__ void gat_edge1_expsum(const long long* __restrict__ ei, long long E, long long Etot,
                                 const float* __restrict__ als, const float* __restrict__ ald,
                                 const float* __restrict__ mf, float* __restrict__ ex1,
                                 float* __restrict__ denom) {
  long long e = (long long)blockIdx.x * blockDim.x + threadIdx.x;
  if (e >= Etot) return;
  long long s, d;
  if (e < E) { s = ei[e]; d = ei[E + e]; } else { s = d = e - E; }
  for (int h = 0; h < 8; ++h) {
    float sc = leaky(als[s * 8 + h] + ald[d * 8 + h]);
    float ex = expf(sc - mf[d * 8 + h]);
    ex1[e * 8 + h] = ex;
    atomicAdd(&denom[d * 8 + h], ex);
  }
}

// ---------- layer-1 aggregation: one block (256 thr) per edge ----------
__global__ void gat_edge1_agg(const long long* __restrict__ ei, long long E,
                              const float* __restrict__ H, const float* __restrict__ ex1,
                              const float* __restrict__ denom, float* __restrict__ agg) {
  long long e = blockIdx.x;
  int t = threadIdx.x;               // 0..255 -> head = t/32, dim = t%32
  long long s, d;
  if (e < E) { s = ei[e]; d = ei[E + e]; } else { s = d = e - E; }
  int h = t >> 5;
  float alpha = ex1[e * 8 + h] / (denom[d * 8 + h] + 1e-16f);
  atomicAdd(&agg[d * 256 + t], alpha * H[s * 256 + t]);
}

// ---------- bias + ELU in place ----------
__global__ void gat_finish1(float* __restrict__ agg, const float* __restrict__ b1, long long n) {
  long long i = (long long)blockIdx.x * blockDim.x + threadIdx.x;
  if (i >= n) return;
  int c = (int)(i & 255);
  float v = agg[i] + b1[c];
  agg[i] = v > 0.f ? v : (expf(v) - 1.f);
}

// ---------- per-node attention logits, layer 2 (1 head x 16 dims) ----------
__global__ void gat_al2(const float* __restrict__ Lp, const float* __restrict__ asrc,
                        const float* __restrict__ adst, float* __restrict__ als,
                        float* __restrict__ ald, long long N) {
  long long n = (long long)blockIdx.x * blockDim.x + threadIdx.x;
  if (n >= N) return;
  const float* p = Lp + n * 16;
  float vs = 0.f, vd = 0.f;
  for (int c = 0; c < 16; ++c) {
    float x = p[c];
    vs += x * asrc[c];
    vd += x * adst[c];
  }
  als[n] = vs;
  ald[n] = vd;
}

__global__ void gat_edge2_max(const long long* __restrict__ ei, long long E, long long Etot,
                              const float* __restrict__ als, const float* __restrict__ ald,
                              unsigned* __restrict__ menc) {
  long long e = (long long)blockIdx.x * blockDim.x + threadIdx.x;
  if (e >= Etot) return;
  long long s, d;
  if (e < E) { s = ei[e]; d = ei[E + e]; } else { s = d = e - E; }
  float sc = leaky(als[s] + ald[d]);
  atomicMax(&menc[d], enc_f32(sc));
}

__global__ void gat_edge2_expsum(const long long* __restrict__ ei, long long E, long long Etot,
                                 const float* __restrict__ als, const float* __restrict__ ald,
                                 const float* __restrict__ mf, float* __restrict__ ex2,
                                 float* __restrict__ denom) {
  long long e = (long long)blockIdx.x * blockDim.x + threadIdx.x;
  if (e >= Etot) return;
  long long s, d;
  if (e < E) { s = ei[e]; d = ei[E + e]; } else { s = d = e - E; }
  float sc = leaky(als[s] + ald[d]);
  float ex = expf(sc - mf[d]);
  ex2[e] = ex;
  atomicAdd(&denom[d], ex);
}

__global__ void gat_edge2_agg(const long long* __restrict__ ei, long long E, long long Etot,
                              const float* __restrict__ Lp, const float* __restrict__ ex2,
                              const float* __restrict__ denom, float* __restrict__ agg2) {
  long long i = (long long)blockIdx.x * blockDim.x + threadIdx.x;
  if (i >= Etot * 16) return;
  long long e = i >> 4;
  int c = (int)(i & 15);
  long long s, d;
  if (e < E) { s = ei[e]; d = ei[E + e]; } else { s = d = e - E; }
  float alpha = ex2[e] / (denom[d] + 1e-16f);
  atomicAdd(&agg2[d * 16 + c], alpha * Lp[s * 16 + c]);
}

// ---------- bias + log_softmax; write (log_softmax, logits) tuple ----------
__global__ void gat_finish2(const float* __restrict__ agg2, const float* __restrict__ b2,
                            float* __restrict__ out, long long N) {
  long long n = (long long)blockIdx.x * blockDim.x + threadIdx.x;
  if (n >= N) return;
  float lg[16];
  float mx = -3.402823466e38f;
  for (int c = 0; c < 16; ++c) {
    float v = agg2[n * 16 + c] + b2[c];
    lg[c] = v;
    mx = v > mx ? v : mx;
  }
  float s = 0.f;
  for (int c = 0; c < 16; ++c) s += expf(lg[c] - mx);
  float lse = mx + logf(s);
  float* lsm = out;
  float* lgo = out + N * 16;
  for (int c = 0; c < 16; ++c) {
    lsm[n * 16 + c] = lg[c] - lse;
    lgo[n * 16 + c] = lg[c];
  }
}

extern "C" void kernel_launch(void* const* d_in, const int* in_sizes, int n_in,
                              void* d_out, int out_size, void* d_ws, size_t ws_size,
                              hipStream_t stream) {
  const float*     x     = (const float*)d_in[0];
  const long long* ei    = (const long long*)d_in[1];
  const float*     W1    = (const float*)d_in[2];
  const float*     asrc1 = (const float*)d_in[3];
  const float*     adst1 = (const float*)d_in[4];
  const float*     b1    = (const float*)d_in[5];
  const float*     W2    = (const float*)d_in[6];
  const float*     asrc2 = (const float*)d_in[7];
  const float*     adst2 = (const float*)d_in[8];
  const float*     b2    = (const float*)d_in[9];

  const long long N    = in_sizes[0] / 256;  // 50000
  const long long E    = in_sizes[1] / 2;    // 800000
  const long long Etot = E + N;              // with self-loops

  // workspace carve-up (floats)
  float* ws = (float*)d_ws;
  float*    H1   = ws;             ws += (size_t)N * 256;
  float*    agg1 = ws;             ws += (size_t)N * 256;
  float*    als1 = ws;             ws += (size_t)N * 8;
  float*    ald1 = ws;             ws += (size_t)N * 8;
  unsigned* m1   = (unsigned*)ws;  ws += (size_t)N * 8;
  float*    den1 = ws;             ws += (size_t)N * 8;
  float*    ex1  = ws;             ws += (size_t)Etot * 8;
  float*    Lp   = ws;             ws += (size_t)N * 16;
  float*    als2 = ws;             ws += (size_t)N;
  float*    ald2 = ws;             ws += (size_t)N;
  unsigned* m2   = (unsigned*)ws;  ws += (size_t)N;
  float*    den2 = ws;             ws += (size_t)N;
  float*    ex2  = ws;             ws += (size_t)Etot;
  float*    agg2 = ws;             ws += (size_t)N * 16;

  const int TB = 256;
  auto blks = [&](long long n) { return (unsigned)((n + TB - 1) / TB); };

  // zero: agg1+als1+ald1+m1+den1 (contiguous), m2+den2, agg2
  long long z1 = N * 256 + 4 * N * 8;
  gat_zero_u32<<<blks(z1), TB, 0, stream>>>((unsigned*)agg1, z1);
  gat_zero_u32<<<blks(2 * N), TB, 0, stream>>>(m2, 2 * N);
  gat_zero_u32<<<blks(N * 16), TB, 0, stream>>>((unsigned*)agg2, N * 16);

  // layer 1
  gat_gemm_wmma_256x256<<<(unsigned)(N / 16), TB, 0, stream>>>(x, W1, H1);
  gat_al1<<<blks(N * 8), TB, 0, stream>>>(H1, asrc1, adst1, als1, ald1, N * 8);
  gat_edge1_max<<<blks(Etot), TB, 0, stream>>>(ei, E, Etot, als1, ald1, m1);
  gat_dec_max<<<blks(N * 8), TB, 0, stream>>>(m1, N * 8);
  gat_edge1_expsum<<<blks(Etot), TB, 0, stream>>>(ei, E, Etot, als1, ald1, (float*)m1, ex1, den1);
  gat_edge1_agg<<<(unsigned)Etot, TB, 0, stream>>>(ei, E, H1, ex1, den1, agg1);
  gat_finish1<<<blks(N * 256), TB, 0, stream>>>(agg1, b1, N * 256);

  // layer 2
  int Mtiles = (int)(N / 16);
  gat_gemm_wmma_256x16<<<(unsigned)((Mtiles + 7) / 8), TB, 0, stream>>>(agg1, W2, Lp, Mtiles);
  gat_al2<<<blks(N), TB, 0, stream>>>(Lp, asrc2, adst2, als2, ald2, N);
  gat_edge2_max<<<blks(Etot), TB, 0, stream>>>(ei, E, Etot, als2, ald2, m2);
  gat_dec_max<<<blks(N), TB, 0, stream>>>(m2, N);
  gat_edge2_expsum<<<blks(Etot), TB, 0, stream>>>(ei, E, Etot, als2, ald2, (float*)m2, ex2, den2);
  gat_edge2_agg<<<blks(Etot * 16), TB, 0, stream>>>(ei, E, Etot, Lp, ex2, den2, agg2);
  gat_finish2<<<blks(N), TB, 0, stream>>>(agg2, b2, (float*)d_out, N);
}